// SpatialAttentionLayer_60876866454290
// MI455X (gfx1250) — compile-verified
//
#include <hip/hip_runtime.h>
#include <hip/hip_bf16.h>

// CDNA5 / gfx1250. Reference collapses analytically (see analysis):
//   out[b,l,i,d] = 4 * (colsum(W)[d] + b[d])   -- X unused, constant over b,l,i.
// Stage 1: colsum(W) via V_WMMA_F32_16X16X4_F32 (ones^T @ W), 1 wave.
// Stage 2: broadcast-fill 16.78 MB of output with b128 stores (HBM-store bound).

typedef __attribute__((ext_vector_type(2))) float v2f;
typedef __attribute__((ext_vector_type(8))) float v8f;

#define D_MODEL 64

// ---------------------------------------------------------------------------
// Stage 1: u[d] = 4 * (sum_k W[k][d] + b[d]), d = 0..63.  One wave32.
// A = ones(16x4)  => D[m,n] = sum_k B[k,n]  (K-lane permutation inside B is
// irrelevant since A is all ones; only N<->lane mapping matters).
// ---------------------------------------------------------------------------
__global__ __launch_bounds__(32) void colsum_w_wmma(const float* __restrict__ W,
                                                    const float* __restrict__ bvec,
                                                    float* __restrict__ u) {
    const int lane = threadIdx.x;          // 0..31
    const int nloc = lane & 15;            // N within tile
    const int khalf = (lane >> 4) * 2;     // 0 or 2: K sub-rows this half-wave holds

    v2f a;
    a.x = 1.0f;                            // all-ones A matrix (16x4 f32)
    a.y = 1.0f;

    for (int tile = 0; tile < 4; ++tile) { // 4 tiles of 16 columns
        const int col = tile * 16 + nloc;
        v8f c = {};
        #pragma unroll
        for (int k0 = 0; k0 < D_MODEL; k0 += 4) {
            v2f bm;
            bm.x = W[(k0 + khalf + 0) * D_MODEL + col];
            bm.y = W[(k0 + khalf + 1) * D_MODEL + col];
            // 8 args: (neg_a, A, neg_b, B, c_mod, C, reuse_a, reuse_b)
            c = __builtin_amdgcn_wmma_f32_16x16x4_f32(
                    false, a, false, bm, (short)0, c, false, false);
        }
        // C/D layout: VGPR0, lanes 0..15 -> D[M=0, N=lane] = colsum(W)[col]
        if (lane < 16) {
            u[tile * 16 + lane] = 4.0f * (c[0] + bvec[tile * 16 + lane]);
        }
    }
}

// ---------------------------------------------------------------------------
// Stage 2: out[idx] = u (broadcast).  Pure store-bandwidth kernel.
// Each thread: one coalesced float4 (b128) store; wave32 writes 512 B lines.
// ---------------------------------------------------------------------------
__global__ __launch_bounds__(256) void broadcast_fill(const float4* __restrict__ u4,
                                                      float4* __restrict__ out,
                                                      int n4) {
    const int idx = blockIdx.x * blockDim.x + threadIdx.x;
    if (idx < n4) {
        out[idx] = u4[idx & 15];           // u = 64 floats = 16 float4s, L1-hot
    }
}

extern "C" void kernel_launch(void* const* d_in, const int* in_sizes, int n_in,
                              void* d_out, int out_size, void* d_ws, size_t ws_size,
                              hipStream_t stream) {
    // setup_inputs order: X, W1, b1, w2, b2, W, b
    const float* W = (const float*)d_in[5];   // (64,64)
    const float* b = (const float*)d_in[6];   // (64,)
    float* u = (float*)d_ws;                  // 64 floats of scratch

    colsum_w_wmma<<<1, 32, 0, stream>>>(W, b, u);

    const int n4 = out_size / 4;              // 4,194,304 floats -> 1,048,576 float4
    const int threads = 256;
    const int blocks = (n4 + threads - 1) / threads;
    broadcast_fill<<<blocks, threads, 0, stream>>>((const float4*)d_ws,
                                                   (float4*)d_out, n4);
}